// DualGAT_7438883357556
// MI455X (gfx1250) — compile-verified
//
#include <hip/hip_runtime.h>
#include <math.h>

#define NN     3072
#define INDIM  32
#define HID    64
#define HEADS  4
#define HD     16
#define SEG    4                 // u-range split for occupancy
#define ULEN   (NN / SEG)        // 768 u's per segment
#define NTILE  (NN / 16)         // 192 v-tiles

typedef __attribute__((ext_vector_type(2))) float v2f;
typedef __attribute__((ext_vector_type(8))) float v8f;

__device__ __forceinline__ float lrelu(float x) { return x > 0.f ? x : 0.2f * x; }

// ---------------------------------------------------------------------------
// Wh = h @ W   (h: N x in_dim, W: in_dim x 64)
// ---------------------------------------------------------------------------
__global__ void linear_k(const float* __restrict__ h, const float* __restrict__ W,
                         float* __restrict__ out, int in_dim) {
  int idx = blockIdx.x * blockDim.x + threadIdx.x;
  if (idx >= NN * HID) return;
  int n = idx >> 6, j = idx & 63;
  const float* hr = h + (size_t)n * in_dim;
  float acc = 0.f;
  for (int k = 0; k < in_dim; ++k) acc = fmaf(hr[k], W[k * HID + j], acc);
  out[idx] = acc;
}

// ---------------------------------------------------------------------------
// s[n][h] = dot(Wh[n][h*16:], a[h][0:16]); t[n][h] = dot(Wh[n][h*16:], a[h][16:32])
// ---------------------------------------------------------------------------
__global__ void st_k(const float* __restrict__ Wh, const float* __restrict__ a,
                     float* __restrict__ s, float* __restrict__ t) {
  int idx = blockIdx.x * blockDim.x + threadIdx.x;
  if (idx >= NN * HEADS) return;
  int n = idx >> 2, h = idx & 3;
  const float* w  = Wh + (size_t)n * HID + h * HD;
  const float* av = a + h * 2 * HD;
  float ss = 0.f, tt = 0.f;
#pragma unroll
  for (int d = 0; d < HD; ++d) {
    ss = fmaf(w[d], av[d], ss);
    tt = fmaf(w[d], av[HD + d], tt);
  }
  s[idx] = ss;
  t[idx] = tt;
}

// ---------------------------------------------------------------------------
// Per-row masked softmax max, all 4 heads in one adjacency sweep.
// One wave32 per row v:  emax[v][h] = lrelu(s[v][h] + max_{adj(v,u)} t[u][h])
// (valid because leaky-relu is monotone increasing)
// ---------------------------------------------------------------------------
__global__ __launch_bounds__(32)
void rowmax_k(const int* __restrict__ adj, const float* __restrict__ s,
              const float* __restrict__ t, float* __restrict__ emax) {
  const int v    = blockIdx.x;
  const int lane = threadIdx.x;
  const int* arow = adj + (size_t)v * NN;
  const float4* t4 = (const float4*)t;
  const float NEG = -__builtin_inff();
  float m0 = NEG, m1 = NEG, m2 = NEG, m3 = NEG;
  for (int u = lane; u < NN; u += 32) {
    const float4 tv = t4[u];
    const bool on = arow[u] != 0;
    m0 = fmaxf(m0, on ? tv.x : NEG);
    m1 = fmaxf(m1, on ? tv.y : NEG);
    m2 = fmaxf(m2, on ? tv.z : NEG);
    m3 = fmaxf(m3, on ? tv.w : NEG);
  }
#pragma unroll
  for (int off = 16; off > 0; off >>= 1) {
    m0 = fmaxf(m0, __shfl_xor(m0, off, 32));
    m1 = fmaxf(m1, __shfl_xor(m1, off, 32));
    m2 = fmaxf(m2, __shfl_xor(m2, off, 32));
    m3 = fmaxf(m3, __shfl_xor(m3, off, 32));
  }
  if (lane == 0) {
    const float* sv = s + (size_t)v * HEADS;
    float* ev = emax + (size_t)v * HEADS;
    ev[0] = lrelu(sv[0] + m0);
    ev[1] = lrelu(sv[1] + m1);
    ev[2] = lrelu(sv[2] + m2);
    ev[3] = lrelu(sv[3] + m3);
  }
}

// ---------------------------------------------------------------------------
// Attention aggregation partials: one wave32 per (16-row v-tile, u-segment).
// All 4 heads per wave; streams u in K=4 chunks through
// V_WMMA_F32_16X16X4_F32 (branch-free: exp masked by multiply).
// Writes unnormalized numerator (16x64) + denominator (16x4) partials.
//   A layout (16x4 f32): lane = M + 16*(k>=2), vgpr = k&1
//   B layout (4x16 f32): lane = N + 16*(k>=2), vgpr = k&1
// ---------------------------------------------------------------------------
__global__ __launch_bounds__(32)
void gat_attn_k(const int* __restrict__ adj, const float* __restrict__ s,
                const float* __restrict__ t, const float* __restrict__ Wh,
                const float* __restrict__ emax,
                float* __restrict__ Pacc, float* __restrict__ Pden) {
  const int tile = blockIdx.x;
  const int seg  = blockIdx.y;
  const int v0   = tile * 16;
  const int useg = seg * ULEN;
  const int lane = threadIdx.x;
  const int m  = lane & 15;       // row M for A, col N for B
  const int kh = lane >> 4;       // K half: k = 2*kh + {0,1}

  const int*    arow = adj + (size_t)(v0 + m) * NN + useg + 2 * kh;
  const float4* t4   = (const float4*)t;
  float sv[HEADS], em[HEADS];
  {
    const float* svp = s + (size_t)(v0 + m) * HEADS;
    const float* emp = emax + (size_t)(v0 + m) * HEADS;
#pragma unroll
    for (int h = 0; h < HEADS; ++h) { sv[h] = svp[h]; em[h] = emp[h]; }
  }

  v8f acc[HEADS] = {v8f{}, v8f{}, v8f{}, v8f{}};
  float den[HEADS] = {0.f, 0.f, 0.f, 0.f};

  for (int u0 = 0; u0 < ULEN; u0 += 4) {
    const int   ub  = useg + u0 + 2 * kh;
    const float mk0 = arow[u0] != 0 ? 1.f : 0.f;      // masks via cndmask
    const float mk1 = arow[u0 + 1] != 0 ? 1.f : 0.f;
    const float4 ta = t4[ub];
    const float4 tb = t4[ub + 1];
    const float taa[HEADS] = {ta.x, ta.y, ta.z, ta.w};
    const float tbb[HEADS] = {tb.x, tb.y, tb.z, tb.w};
    const float* w0 = Wh + (size_t)ub * HID + m;
    const float* w1 = w0 + HID;
#pragma unroll
    for (int h = 0; h < HEADS; ++h) {
      const float e0 = mk0 * __expf(lrelu(sv[h] + taa[h]) - em[h]);
      const float e1 = mk1 * __expf(lrelu(sv[h] + tbb[h]) - em[h]);
      den[h] += e0 + e1;
      v2f A; A.x = e0;        A.y = e1;
      v2f B; B.x = w0[h * HD]; B.y = w1[h * HD];
      acc[h] = __builtin_amdgcn_wmma_f32_16x16x4_f32(
          false, A, false, B, (short)0, acc[h], false, false);
    }
  }

  // ---- write partials; C/D layout: vgpr j -> M = j + 8*kh, N = lane&15 ----
  float* pa = Pacc + ((size_t)(tile * SEG + seg) * 16) * HID;
  float* pd = Pden + ((size_t)(tile * SEG + seg) * 16) * HEADS;
#pragma unroll
  for (int h = 0; h < HEADS; ++h) {
    const float d = den[h] + __shfl_xor(den[h], 16, 32);
    if (lane < 16) pd[m * HEADS + h] = d;
#pragma unroll
    for (int j = 0; j < 8; ++j) {
      const int M = j + 8 * kh;
      pa[M * HID + h * HD + m] = acc[h][j];
    }
  }
}

// ---------------------------------------------------------------------------
// Combine u-segment partials: normalize + ELU -> Hout (N x 64)
// ---------------------------------------------------------------------------
__global__ void combine_k(const float* __restrict__ Pacc,
                          const float* __restrict__ Pden,
                          float* __restrict__ Hout) {
  int idx = blockIdx.x * blockDim.x + threadIdx.x;
  if (idx >= NN * HID) return;
  const int n = idx >> 6, c = idx & 63;
  const int tile = n >> 4, M = n & 15, h = c >> 4;
  float num = 0.f, dd = 0.f;
#pragma unroll
  for (int seg = 0; seg < SEG; ++seg) {
    num += Pacc[((size_t)(tile * SEG + seg) * 16 + M) * HID + c];
    dd  += Pden[((size_t)(tile * SEG + seg) * 16 + M) * HEADS + h];
  }
  float val = num / dd;
  val = val > 0.f ? val : (__expf(val) - 1.f);   // ELU
  Hout[idx] = val;
}

// ---------------------------------------------------------------------------
// Semantic attention fuse: betas = softmax([Hi.qi, Hc.qc]); H = b0*Hi + b1*Hc
// ---------------------------------------------------------------------------
__global__ void fuse_k(const float* __restrict__ Hi, const float* __restrict__ Hc,
                       const float* __restrict__ qi, const float* __restrict__ qc,
                       float* __restrict__ Hout) {
  int n = blockIdx.x * blockDim.x + threadIdx.x;
  if (n >= NN) return;
  const float* hi = Hi + (size_t)n * HID;
  const float* hc = Hc + (size_t)n * HID;
  float ai = 0.f, ac = 0.f;
#pragma unroll
  for (int d = 0; d < HID; ++d) {
    ai = fmaf(hi[d], qi[d], ai);
    ac = fmaf(hc[d], qc[d], ac);
  }
  float mx = fmaxf(ai, ac);
  float ei = __expf(ai - mx), ec = __expf(ac - mx);
  float inv = 1.f / (ei + ec);
  float b0 = ei * inv, b1 = ec * inv;
  float* ho = Hout + (size_t)n * HID;
#pragma unroll
  for (int d = 0; d < HID; ++d) ho[d] = b0 * hi[d] + b1 * hc[d];
}

// ---------------------------------------------------------------------------
// out = relu(H @ w1 + b1) @ w2 + b2    (64 -> 32 -> 1)
// ---------------------------------------------------------------------------
__global__ void mlp_k(const float* __restrict__ H, const float* __restrict__ w1,
                      const float* __restrict__ b1, const float* __restrict__ w2,
                      const float* __restrict__ b2, float* __restrict__ out) {
  int n = blockIdx.x * blockDim.x + threadIdx.x;
  if (n >= NN) return;
  const float* hr = H + (size_t)n * HID;
  float o = b2[0];
  for (int j = 0; j < HID / 2; ++j) {
    float acc = b1[j];
    for (int k = 0; k < HID; ++k) acc = fmaf(hr[k], w1[k * (HID / 2) + j], acc);
    acc = acc > 0.f ? acc : 0.f;
    o = fmaf(acc, w2[j], o);
  }
  out[n] = o;
}

// ---------------------------------------------------------------------------
extern "C" void kernel_launch(void* const* d_in, const int* in_sizes, int n_in,
                              void* d_out, int out_size, void* d_ws, size_t ws_size,
                              hipStream_t stream) {
  const float* x       = (const float*)d_in[0];
  const int*   adj_ind = (const int*)d_in[1];
  const int*   adj_cor = (const int*)d_in[2];
  const float* W1i = (const float*)d_in[3];
  const float* a1i = (const float*)d_in[4];
  const float* W1c = (const float*)d_in[5];
  const float* a1c = (const float*)d_in[6];
  const float* q1i = (const float*)d_in[7];
  const float* q1c = (const float*)d_in[8];
  const float* W2i = (const float*)d_in[9];
  const float* a2i = (const float*)d_in[10];
  const float* W2c = (const float*)d_in[11];
  const float* a2c = (const float*)d_in[12];
  const float* q2i = (const float*)d_in[13];
  const float* q2c = (const float*)d_in[14];
  const float* mw1 = (const float*)d_in[15];
  const float* mb1 = (const float*)d_in[16];
  const float* mw2 = (const float*)d_in[17];
  const float* mb2 = (const float*)d_in[18];
  float* out = (float*)d_out;

  float* ws   = (float*)d_ws;
  float* Wh_i = ws;                        // N*64
  float* Wh_c = Wh_i + NN * HID;           // N*64
  float* Hi   = Wh_c + NN * HID;           // N*64
  float* Hc   = Hi + NN * HID;             // N*64
  float* H1   = Hc + NN * HID;             // N*64
  float* H2   = H1 + NN * HID;             // N*64
  float* s_i  = H2 + NN * HID;             // N*4
  float* t_i  = s_i + NN * HEADS;
  float* s_c  = t_i + NN * HEADS;
  float* t_c  = s_c + NN * HEADS;
  float* em_i = t_c + NN * HEADS;          // N*4
  float* em_c = em_i + NN * HEADS;         // N*4
  float* Pacc = em_c + NN * HEADS;         // NTILE*SEG*16*64
  float* Pden = Pacc + (size_t)NTILE * SEG * 16 * HID;  // NTILE*SEG*16*4

  dim3 b256(256), b32(32);
  dim3 gLin((NN * HID + 255) / 256);
  dim3 gST((NN * HEADS + 255) / 256);
  dim3 gRow(NN);
  dim3 gAttn(NTILE, SEG);
  dim3 gN((NN + 255) / 256);

  // ---- layer 1 ----
  linear_k<<<gLin, b256, 0, stream>>>(x, W1i, Wh_i, INDIM);
  linear_k<<<gLin, b256, 0, stream>>>(x, W1c, Wh_c, INDIM);
  st_k<<<gST, b256, 0, stream>>>(Wh_i, a1i, s_i, t_i);
  st_k<<<gST, b256, 0, stream>>>(Wh_c, a1c, s_c, t_c);
  rowmax_k<<<gRow, b32, 0, stream>>>(adj_ind, s_i, t_i, em_i);
  gat_attn_k<<<gAttn, b32, 0, stream>>>(adj_ind, s_i, t_i, Wh_i, em_i, Pacc, Pden);
  combine_k<<<gLin, b256, 0, stream>>>(Pacc, Pden, Hi);
  rowmax_k<<<gRow, b32, 0, stream>>>(adj_cor, s_c, t_c, em_c);
  gat_attn_k<<<gAttn, b32, 0, stream>>>(adj_cor, s_c, t_c, Wh_c, em_c, Pacc, Pden);
  combine_k<<<gLin, b256, 0, stream>>>(Pacc, Pden, Hc);
  fuse_k<<<gN, b256, 0, stream>>>(Hi, Hc, q1i, q1c, H1);

  // ---- layer 2 ----
  linear_k<<<gLin, b256, 0, stream>>>(H1, W2i, Wh_i, HID);
  linear_k<<<gLin, b256, 0, stream>>>(H1, W2c, Wh_c, HID);
  st_k<<<gST, b256, 0, stream>>>(Wh_i, a2i, s_i, t_i);
  st_k<<<gST, b256, 0, stream>>>(Wh_c, a2c, s_c, t_c);
  rowmax_k<<<gRow, b32, 0, stream>>>(adj_ind, s_i, t_i, em_i);
  gat_attn_k<<<gAttn, b32, 0, stream>>>(adj_ind, s_i, t_i, Wh_i, em_i, Pacc, Pden);
  combine_k<<<gLin, b256, 0, stream>>>(Pacc, Pden, Hi);
  rowmax_k<<<gRow, b32, 0, stream>>>(adj_cor, s_c, t_c, em_c);
  gat_attn_k<<<gAttn, b32, 0, stream>>>(adj_cor, s_c, t_c, Wh_c, em_c, Pacc, Pden);
  combine_k<<<gLin, b256, 0, stream>>>(Pacc, Pden, Hc);
  fuse_k<<<gN, b256, 0, stream>>>(Hi, Hc, q2i, q2c, H2);

  // ---- MLP head ----
  mlp_k<<<gN, b256, 0, stream>>>(H2, mw1, mb1, mw2, mb2, out);
}